// TopoFlowGNN_20538533609734
// MI455X (gfx1250) — compile-verified
//
#include <hip/hip_runtime.h>
#include <hip/hip_bf16.h>
#include <math.h>

typedef float v2f __attribute__((ext_vector_type(2)));
typedef float v8f __attribute__((ext_vector_type(8)));

#define NEG_SLOPE 0.2f
#define ORD_NEGINF 0x007fffffu  /* order-preserving encoding of -inf */
#define GEMM_NC 64              /* both WMMA GEMMs have 64 output channels */
#define GEMM_KMAX 128

// ---------- order-preserving float<->uint map for atomicMax on floats ----------
__device__ __forceinline__ unsigned f2ord(float f) {
    unsigned u = __float_as_uint(f);
    return (u & 0x80000000u) ? ~u : (u | 0x80000000u);
}
__device__ __forceinline__ float ord2f(unsigned u) {
    return (u & 0x80000000u) ? __uint_as_float(u & 0x7fffffffu) : __uint_as_float(~u);
}

// ---------- fills ----------
__global__ void fill_f32(float* p, float v, int n) {
    int i = blockIdx.x * blockDim.x + threadIdx.x;
    if (i < n) p[i] = v;
}
__global__ void fill_f32_v4(float4* p, float v, int n4) {
    int i = blockIdx.x * blockDim.x + threadIdx.x;
    if (i < n4) p[i] = make_float4(v, v, v, v);
}
__global__ void fill_u32(unsigned* p, unsigned v, int n) {
    int i = blockIdx.x * blockDim.x + threadIdx.x;
    if (i < n) p[i] = v;
}

// ---------- layer 1: h = x@W1 (K=2), plus attention logits (H=2, C=64) ----------
__global__ void layer1_node(const float* __restrict__ x, const float* __restrict__ w1,
                            const float* __restrict__ as1, const float* __restrict__ ad1,
                            float* __restrict__ h, float* __restrict__ alsrc,
                            float* __restrict__ aldst, int N) {
    int n = blockIdx.x * blockDim.x + threadIdx.x;
    if (n >= N) return;
    float x0 = x[2 * n], x1 = x[2 * n + 1];
    float ss0 = 0.f, sd0 = 0.f, ss1 = 0.f, sd1 = 0.f;
    size_t base = (size_t)n * 128;
    for (int c = 0; c < 128; ++c) {
        float v = x0 * w1[c] + x1 * w1[128 + c];
        h[base + c] = v;
        int hd = c >> 6, cc = c & 63;
        float as = as1[hd * 64 + cc];
        float ad = ad1[hd * 64 + cc];
        if (hd == 0) { ss0 += v * as; sd0 += v * ad; }
        else         { ss1 += v * as; sd1 += v * ad; }
    }
    alsrc[2 * n] = ss0; alsrc[2 * n + 1] = ss1;
    aldst[2 * n] = sd0; aldst[2 * n + 1] = sd1;
}

// ---------- attention logits (general H, C; C multiple of 4) ----------
__global__ void attn_logits(const float* __restrict__ h, const float* __restrict__ a_src,
                            const float* __restrict__ a_dst, float* __restrict__ alsrc,
                            float* __restrict__ aldst, int N, int H, int C) {
    int i = blockIdx.x * blockDim.x + threadIdx.x;
    if (i >= N * H) return;
    int n = i / H, hd = i % H;
    const float4* hp = (const float4*)(h + (size_t)n * H * C + (size_t)hd * C);
    const float4* asp = (const float4*)(a_src + hd * C);
    const float4* adp = (const float4*)(a_dst + hd * C);
    float ss = 0.f, sd = 0.f;
    for (int c4 = 0; c4 < (C >> 2); ++c4) {
        float4 v = hp[c4], a = asp[c4], b = adp[c4];
        ss += v.x * a.x + v.y * a.y + v.z * a.z + v.w * a.w;
        sd += v.x * b.x + v.y * b.y + v.z * b.z + v.w * b.w;
    }
    alsrc[i] = ss;
    aldst[i] = sd;
}

// ---------- edge pass 1: e = lrelu(al_src[src]+al_dst[dst]); segment max over dst ----------
__global__ void edge_max(const float* __restrict__ alsrc, const float* __restrict__ aldst,
                         const int* __restrict__ esrc, const int* __restrict__ edst,
                         float* __restrict__ ebuf, unsigned* __restrict__ mbuf,
                         int E, int N, int H) {
    int i = blockIdx.x * blockDim.x + threadIdx.x;
    int ET = E + N;
    if (i >= ET * H) return;
    int e = i / H, hd = i - e * H;
    int s, d;
    if (e < E) { s = esrc[e]; d = edst[e]; } else { s = e - E; d = s; }
    float v = alsrc[s * H + hd] + aldst[d * H + hd];
    v = v > 0.f ? v : NEG_SLOPE * v;
    ebuf[i] = v;
    atomicMax(&mbuf[d * H + hd], f2ord(v));
}

// ---------- edge pass 2: e = exp(e - m[dst]); segment sum over dst ----------
__global__ void edge_expsum(const int* __restrict__ esrc, const int* __restrict__ edst,
                            float* __restrict__ ebuf, const unsigned* __restrict__ mbuf,
                            float* __restrict__ sbuf, int E, int N, int H) {
    int i = blockIdx.x * blockDim.x + threadIdx.x;
    int ET = E + N;
    if (i >= ET * H) return;
    int e = i / H, hd = i - e * H;
    int d = (e < E) ? edst[e] : (e - E);
    float mm = ord2f(mbuf[d * H + hd]);
    float ex = __expf(ebuf[i] - mm);
    ebuf[i] = ex;
    atomicAdd(&sbuf[d * H + hd], ex);
}

// ---------- edge pass 3 (float4): agg[dst] += h[src] * alpha ----------
__global__ void edge_scatter_v4(const int* __restrict__ esrc, const int* __restrict__ edst,
                                const float* __restrict__ ebuf, const float* __restrict__ sbuf,
                                const float* __restrict__ h, float* __restrict__ agg,
                                int E, int N, int H, int C) {
    int HC = H * C;
    int QC = HC >> 2;                         // float4 groups per edge
    long long i = (long long)blockIdx.x * blockDim.x + threadIdx.x;
    long long total = (long long)(E + N) * QC;
    if (i >= total) return;
    int e = (int)(i / QC);
    int q = (int)(i - (long long)e * QC);
    int c0 = q << 2;
    int hd = c0 / C;                          // C multiple of 4: whole group in one head
    int s, d;
    if (e < E) { s = esrc[e]; d = edst[e]; } else { s = e - E; d = s; }
    float alpha = ebuf[(size_t)e * H + hd] / (sbuf[d * H + hd] + 1e-16f);
    float4 hv = *(const float4*)(h + (size_t)s * HC + c0);
    float* ap = agg + (size_t)d * HC + c0;
    atomicAdd(ap + 0, hv.x * alpha);
    atomicAdd(ap + 1, hv.y * alpha);
    atomicAdd(ap + 2, hv.z * alpha);
    atomicAdd(ap + 3, hv.w * alpha);
}

// ---------- finalize: h = elu(agg + bias), in place ----------
__global__ void finalize_elu(float* __restrict__ agg, const float* __restrict__ bias,
                             int N, int HC) {
    long long i = (long long)blockIdx.x * blockDim.x + threadIdx.x;
    if (i >= (long long)N * HC) return;
    float v = agg[i] + bias[(int)(i % HC)];
    agg[i] = v > 0.f ? v : (__expf(v) - 1.f);
}

// ---------- WMMA f32 GEMM: Out[M,64] = A[M,K] @ W[K,64]; M%16==0, K%4==0, K<=128 ----------
// One wave computes a 16x64 strip (4 accumulators, A fragment reused 4x per K-step).
// W is staged in LDS once per block (b128 cooperative load), B fragments come from LDS.
__global__ void gemm_wmma_f32(const float* __restrict__ A, const float* __restrict__ W,
                              float* __restrict__ Out, int M, int K) {
    __shared__ float ldsW[GEMM_KMAX * GEMM_NC];   // 32 KB max

    // cooperative stage of W[K x 64] into LDS
    {
        const float4* W4 = (const float4*)W;
        float4* L4 = (float4*)ldsW;
        int n4 = K * (GEMM_NC / 4);
        for (int idx = threadIdx.x; idx < n4; idx += blockDim.x) L4[idx] = W4[idx];
    }
    __syncthreads();

    int lane = threadIdx.x & 31;
    int widx = threadIdx.x >> 5;
    int mt = blockIdx.x * (blockDim.x >> 5) + widx;
    int mtiles = M >> 4;
    if (mt >= mtiles) return;                 // wave-uniform: EXEC all-ones for WMMA

    int half = lane >> 4;                     // 0: lanes 0-15, 1: lanes 16-31
    int l16 = lane & 15;
    const float* Arow = A + (size_t)((mt << 4) + l16) * K;

    v8f acc0 = {0.f,0.f,0.f,0.f,0.f,0.f,0.f,0.f};
    v8f acc1 = acc0, acc2 = acc0, acc3 = acc0;

    for (int k0 = 0; k0 < K; k0 += 4) {
        int ka = k0 + half * 2;               // ISA A layout: lanes 16-31 hold K=k0+2,k0+3
        v2f a;
        a[0] = Arow[ka];
        a[1] = Arow[ka + 1];
        const float* Wk0 = ldsW + ka * GEMM_NC + l16;
        const float* Wk1 = Wk0 + GEMM_NC;
        v2f b;
        b[0] = Wk0[0];  b[1] = Wk1[0];
        acc0 = __builtin_amdgcn_wmma_f32_16x16x4_f32(false, a, false, b, (short)0, acc0, false, false);
        b[0] = Wk0[16]; b[1] = Wk1[16];
        acc1 = __builtin_amdgcn_wmma_f32_16x16x4_f32(false, a, false, b, (short)0, acc1, false, false);
        b[0] = Wk0[32]; b[1] = Wk1[32];
        acc2 = __builtin_amdgcn_wmma_f32_16x16x4_f32(false, a, false, b, (short)0, acc2, false, false);
        b[0] = Wk0[48]; b[1] = Wk1[48];
        acc3 = __builtin_amdgcn_wmma_f32_16x16x4_f32(false, a, false, b, (short)0, acc3, false, false);
    }

    int orow = (mt << 4) + half * 8;          // C/D: VGPR r -> row r + 8*half
    for (int r = 0; r < 8; ++r) {
        float* o = Out + (size_t)(orow + r) * GEMM_NC + l16;
        o[0]  = acc0[r];
        o[16] = acc1[r];
        o[32] = acc2[r];
        o[48] = acc3[r];
    }
}

// ---------- pooling ----------
__global__ void pool_count(const int* __restrict__ batch, float* __restrict__ cnts, int N) {
    int n = blockIdx.x * blockDim.x + threadIdx.x;
    if (n >= N) return;
    atomicAdd(&cnts[batch[n]], 1.0f);
}
__global__ void pool_sum(const int* __restrict__ batch, const float* __restrict__ h,
                         float* __restrict__ sums, int N) {
    long long i = (long long)blockIdx.x * blockDim.x + threadIdx.x;
    if (i >= (long long)N * 64) return;
    int n = (int)(i >> 6), c = (int)(i & 63);
    atomicAdd(&sums[batch[n] * 64 + c], h[(size_t)n * 64 + c]);
}

// ---------- MLP head ----------
__global__ void mlp_head(const float* __restrict__ sums, const float* __restrict__ cnts,
                         const float* __restrict__ fw1, const float* __restrict__ fb1,
                         const float* __restrict__ fw2, const float* __restrict__ fb2,
                         float* __restrict__ out, int G) {
    int g = threadIdx.x;
    if (g >= G) return;
    float cnt = fmaxf(cnts[g], 1.0f);
    float feat[64];
    for (int c = 0; c < 64; ++c) feat[c] = sums[g * 64 + c] / cnt;
    float o = fb2[0];
    for (int j = 0; j < 32; ++j) {
        float a = fb1[j];
        for (int c = 0; c < 64; ++c) a += feat[c] * fw1[c * 32 + j];
        a = fmaxf(a, 0.f);
        o += a * fw2[j];
    }
    out[g] = o;
}

static inline int nblk(long long total, int bs) { return (int)((total + bs - 1) / bs); }

extern "C" void kernel_launch(void* const* d_in, const int* in_sizes, int n_in,
                              void* d_out, int out_size, void* d_ws, size_t ws_size,
                              hipStream_t stream) {
    const float* x      = (const float*)d_in[0];
    const int*   ei     = (const int*)d_in[1];
    const int*   batch  = (const int*)d_in[2];
    const float* w1     = (const float*)d_in[3];
    const float* a_src1 = (const float*)d_in[4];
    const float* a_dst1 = (const float*)d_in[5];
    const float* b1     = (const float*)d_in[6];
    const float* w2     = (const float*)d_in[7];
    const float* a_src2 = (const float*)d_in[8];
    const float* a_dst2 = (const float*)d_in[9];
    const float* b2     = (const float*)d_in[10];
    const float* w3     = (const float*)d_in[11];
    const float* a_src3 = (const float*)d_in[12];
    const float* a_dst3 = (const float*)d_in[13];
    const float* b3     = (const float*)d_in[14];
    const float* fw1    = (const float*)d_in[15];
    const float* fb1    = (const float*)d_in[16];
    const float* fw2    = (const float*)d_in[17];
    const float* fb2    = (const float*)d_in[18];
    float* out = (float*)d_out;

    const int N  = in_sizes[0] / 2;       // 50000
    const int E  = in_sizes[1] / 2;       // 800000
    const int G  = 64;
    const int ET = E + N;
    const int* esrc = ei;
    const int* edst = ei + E;

    // workspace carve-out (~60 MB total)
    char* ws = (char*)d_ws;
    size_t off = 0;
    auto carve = [&](size_t bytes) -> char* {
        char* p = ws + off;
        off = (off + bytes + 255) & ~(size_t)255;
        return p;
    };
    float*    buf0  = (float*)   carve((size_t)N * 128 * 4); // linear h of current layer
    float*    buf1  = (float*)   carve((size_t)N * 128 * 4); // aggregation / layer output
    float*    alsrc = (float*)   carve((size_t)N * 2 * 4);
    float*    aldst = (float*)   carve((size_t)N * 2 * 4);
    unsigned* mbuf  = (unsigned*)carve((size_t)N * 2 * 4);
    float*    sbuf  = (float*)   carve((size_t)N * 2 * 4);
    float*    ebuf  = (float*)   carve((size_t)ET * 2 * 4);
    float*    sums  = (float*)   carve((size_t)G * 64 * 4);
    float*    cnts  = (float*)   carve((size_t)G * 4);

    const int BS = 256;

    auto run_edges = [&](const float* h, float* agg, int H, int C) {
        int HC = H * C;
        fill_f32_v4<<<nblk((long long)N * HC / 4, BS), BS, 0, stream>>>((float4*)agg, 0.f,
                                                                        N * HC / 4);
        fill_f32<<<nblk(N * H, BS), BS, 0, stream>>>(sbuf, 0.f, N * H);
        fill_u32<<<nblk(N * H, BS), BS, 0, stream>>>(mbuf, ORD_NEGINF, N * H);
        edge_max<<<nblk((long long)ET * H, BS), BS, 0, stream>>>(alsrc, aldst, esrc, edst,
                                                                 ebuf, mbuf, E, N, H);
        edge_expsum<<<nblk((long long)ET * H, BS), BS, 0, stream>>>(esrc, edst, ebuf, mbuf,
                                                                    sbuf, E, N, H);
        edge_scatter_v4<<<nblk((long long)ET * (HC / 4), BS), BS, 0, stream>>>(
            esrc, edst, ebuf, sbuf, h, agg, E, N, H, C);
    };

    // ---- layer 1: 2 -> 128 (H=2, C=64) ----
    layer1_node<<<nblk(N, BS), BS, 0, stream>>>(x, w1, a_src1, a_dst1, buf0, alsrc, aldst, N);
    run_edges(buf0, buf1, 2, 64);
    finalize_elu<<<nblk((long long)N * 128, BS), BS, 0, stream>>>(buf1, b1, N, 128);

    // ---- layer 2: 128 -> 64 (H=1) : WMMA GEMM buf0 = buf1 @ w2 ----
    {
        int mtiles = N / 16;                          // 3125
        gemm_wmma_f32<<<nblk(mtiles, 8), BS, 0, stream>>>(buf1, w2, buf0, N, 128);
    }
    attn_logits<<<nblk(N, BS), BS, 0, stream>>>(buf0, a_src2, a_dst2, alsrc, aldst, N, 1, 64);
    run_edges(buf0, buf1, 1, 64);
    finalize_elu<<<nblk((long long)N * 64, BS), BS, 0, stream>>>(buf1, b2, N, 64);

    // ---- layer 3: 64 -> 64 (H=1) : WMMA GEMM buf0 = buf1 @ w3 ----
    {
        int mtiles = N / 16;
        gemm_wmma_f32<<<nblk(mtiles, 8), BS, 0, stream>>>(buf1, w3, buf0, N, 64);
    }
    attn_logits<<<nblk(N, BS), BS, 0, stream>>>(buf0, a_src3, a_dst3, alsrc, aldst, N, 1, 64);
    run_edges(buf0, buf1, 1, 64);
    finalize_elu<<<nblk((long long)N * 64, BS), BS, 0, stream>>>(buf1, b3, N, 64);

    // ---- global mean pool + MLP head ----
    fill_f32<<<nblk(G * 64, BS), BS, 0, stream>>>(sums, 0.f, G * 64);
    fill_f32<<<1, BS, 0, stream>>>(cnts, 0.f, G);
    pool_count<<<nblk(N, BS), BS, 0, stream>>>(batch, cnts, N);
    pool_sum<<<nblk((long long)N * 64, BS), BS, 0, stream>>>(batch, buf1, sums, N);
    mlp_head<<<1, 64, 0, stream>>>(sums, cnts, fw1, fb1, fw2, fb2, out, G);
}